// ScaledDotProductTAtt_17145509446375
// MI455X (gfx1250) — compile-verified
//
#include <hip/hip_runtime.h>

typedef float v2f __attribute__((ext_vector_type(2)));
typedef float v8f __attribute__((ext_vector_type(8)));

// Fixed reference shapes
constexpr int Bn = 16, Nn = 325, Hn = 8, Ln = 24, DKn = 64, DVn = 64;
constexpr int TILES     = Bn * Nn * Hn;        // 41600 (b,n,h) tiles
constexpr int QK_STRIDE = Ln * DKn;            // 1536 floats per Q/K/V tile
constexpr int SS        = Ln * Ln;             // 576 floats per score tile
constexpr long long CTX_ELEMS = (long long)TILES * Ln * DVn;

__global__ __launch_bounds__(32) void tatt_wmma_kernel(
    const float* __restrict__ Q, const float* __restrict__ K,
    const float* __restrict__ V, const float* __restrict__ mask,
    const float* __restrict__ res,
    float* __restrict__ ctx_out, float* __restrict__ sc_out)
{
    __shared__ float Qs[QK_STRIDE];
    __shared__ float Ks[QK_STRIDE];
    __shared__ float Sl[SS];

    const int tile = blockIdx.x;
    const int lane = (int)threadIdx.x;   // wave32: one wave per block
    const int half = lane >> 4;          // 0: lanes 0-15, 1: lanes 16-31
    const int lm   = lane & 15;

    const float* Qg = Q    + (size_t)tile * QK_STRIDE;
    const float* Kg = K    + (size_t)tile * QK_STRIDE;
    const float* Vg = V    + (size_t)tile * QK_STRIDE;
    const float* Mg = mask + (size_t)tile * SS;
    const float* Rg = res  + (size_t)tile * SS;
    float*       Cg = ctx_out + (size_t)tile * (Ln * DVn);
    float*       Sg = sc_out  + (size_t)tile * SS;

    // ---- Stage Q, K tiles into LDS (coalesced: 32 consecutive floats/iter) ----
    for (int i = lane; i < QK_STRIDE; i += 32) {
        Qs[i] = Qg[i];
        Ks[i] = Kg[i];
    }
    __syncthreads();

    // ---- GEMM 1: S = Q * K^T * (1/sqrt(64)) + res, then masked_fill ----
    // Output 24x24 padded to 2x2 tiles of 16x16; K-dim = 64 -> 16 steps of 4.
    for (int mi = 0; mi < 2; ++mi) {
        const int  arow = mi * 16 + lm;
        const bool rowv = arow < Ln;
        for (int ni = 0; ni < 2; ++ni) {
            const int  bcol = ni * 16 + lm;     // B = K^T, so B[kk][n] = K[n][kk]
            const bool colv = bcol < Ln;
            v8f acc = {};
#pragma unroll
            for (int k0 = 0; k0 < DKn; k0 += 4) {
                const int kk = k0 + 2 * half;   // A/B VGPR j holds K = j + 2*half
                v2f a, b;
                a.x = rowv ? Qs[arow * DKn + kk]     : 0.0f;
                a.y = rowv ? Qs[arow * DKn + kk + 1] : 0.0f;
                b.x = colv ? Ks[bcol * DKn + kk]     : 0.0f;
                b.y = colv ? Ks[bcol * DKn + kk + 1] : 0.0f;
                acc = __builtin_amdgcn_wmma_f32_16x16x4_f32(
                    false, a, false, b, (short)0, acc, false, false);
            }
            // C/D layout: VGPR r -> row = r + 8*half, col = lane%16
#pragma unroll
            for (int r = 0; r < 8; ++r) {
                const int orow = mi * 16 + r + 8 * half;
                const int ocol = ni * 16 + lm;
                if (orow < Ln && ocol < Ln) {
                    float s = acc[r] * 0.125f + Rg[orow * Ln + ocol];
                    if (Mg[orow * Ln + ocol] > 0.5f) s = -1.0e9f;
                    Sg[orow * Ln + ocol] = s;   // scores output (2nd tuple element)
                    Sl[orow * Ln + ocol] = s;   // keep in LDS for softmax
                }
            }
        }
    }
    __syncthreads();

    // ---- Softmax over axis 3 == the q axis: per-COLUMN softmax of S ----
    // Lane c (0..23) owns column c; loop over the 24 rows.
    if (lane < Ln) {
        float m = -3.402823466e38f;
        for (int r = 0; r < Ln; ++r) m = fmaxf(m, Sl[r * Ln + lane]);
        float sum = 0.0f;
        for (int r = 0; r < Ln; ++r) {
            float e = __expf(Sl[r * Ln + lane] - m);
            Sl[r * Ln + lane] = e;
            sum += e;
        }
        const float inv = 1.0f / sum;
        for (int r = 0; r < Ln; ++r) Sl[r * Ln + lane] *= inv;
    }
    __syncthreads();

    // ---- GEMM 2: context = P(24x24) * V(24x64); K-dim = 24 -> 6 steps of 4 ----
    for (int mi = 0; mi < 2; ++mi) {
        const int  arow = mi * 16 + lm;
        const bool rowv = arow < Ln;
        for (int nj = 0; nj < 4; ++nj) {
            v8f acc = {};
#pragma unroll
            for (int k0 = 0; k0 < Ln; k0 += 4) {
                const int kk = k0 + 2 * half;           // kk <= 23, always valid
                v2f a, b;
                a.x = rowv ? Sl[arow * Ln + kk]     : 0.0f;
                a.y = rowv ? Sl[arow * Ln + kk + 1] : 0.0f;
                // B = V: lane-contiguous gather straight from global (coalesced 64B/half-wave)
                b.x = Vg[kk       * DVn + nj * 16 + lm];
                b.y = Vg[(kk + 1) * DVn + nj * 16 + lm];
                acc = __builtin_amdgcn_wmma_f32_16x16x4_f32(
                    false, a, false, b, (short)0, acc, false, false);
            }
#pragma unroll
            for (int r = 0; r < 8; ++r) {
                const int orow = mi * 16 + r + 8 * half;
                if (orow < Ln) Cg[orow * DVn + nj * 16 + lm] = acc[r];
            }
        }
    }
}

extern "C" void kernel_launch(void* const* d_in, const int* in_sizes, int n_in,
                              void* d_out, int out_size, void* d_ws, size_t ws_size,
                              hipStream_t stream) {
    const float* Q    = (const float*)d_in[0];
    const float* K    = (const float*)d_in[1];
    const float* V    = (const float*)d_in[2];
    const float* mask = (const float*)d_in[3];
    const float* res  = (const float*)d_in[4];
    float* out = (float*)d_out;

    // Tuple return order: (context, scores) -> context first, then scores.
    float* ctx_out = out;
    float* sc_out  = out + CTX_ELEMS;

    tatt_wmma_kernel<<<TILES, 32, 0, stream>>>(Q, K, V, mask, res, ctx_out, sc_out);
}